// AugLUT_36455682408915
// MI455X (gfx1250) — compile-verified
//
#include <hip/hip_runtime.h>
#include <cstdint>

#define N_BINS   20
#define EPS      1e-5f
#define THREADS  256
#define ITERS    8

typedef float v4f __attribute__((ext_vector_type(4)));

// out = y0[idx] + slope[idx] * (x - idx/19), idx = clamp(floor(x*19), 0, 18)
__device__ __forceinline__ float apply1(float xv, const float2* __restrict__ lutp)
{
    constexpr float kBins = (float)(N_BINS - 1);       // 19
    constexpr float kInv  = 1.0f / (float)(N_BINS - 1);
    int idx = (int)(xv * kBins);                       // x >= 0 -> trunc == floor
    idx = idx < 0 ? 0 : (idx > (N_BINS - 2) ? (N_BINS - 2) : idx);
    float2 e = lutp[idx];                              // ds_load_b64 {y0, slope}
    float x0 = (float)idx * kInv;
    return fmaf(e.y, xv - x0, e.x);
}

__device__ __forceinline__ v4f apply4(v4f v, const float2* __restrict__ lutp)
{
    v4f r;
    r.x = apply1(v.x, lutp);
    r.y = apply1(v.y, lutp);
    r.z = apply1(v.z, lutp);
    r.w = apply1(v.w, lutp);
    return r;
}

__global__ __launch_bounds__(THREADS)
void AugLUT_36455682408915_kernel(const float* __restrict__ x,
                                  const float* __restrict__ ran_y,
                                  float* __restrict__ out,
                                  int vol4)  // float4s per batch
{
    __shared__ float  rawY[N_BINS];
    __shared__ float2 lut[N_BINS - 1];

    const int b   = blockIdx.y;
    const int tid = threadIdx.x;

    // ---- Stage this batch's 20 LUT inputs into LDS via CDNA5 async copy ----
    if (tid < N_BINS) {
        const float* gsrc = ran_y + (size_t)b * N_BINS + tid;
        uint32_t lds_off = (uint32_t)(uintptr_t)
            (__attribute__((address_space(3))) float*)(&rawY[tid]);
        asm volatile("global_load_async_to_lds_b32 %0, %1, off"
                     :
                     : "v"(lds_off), "v"(gsrc)
                     : "memory");
    }
    asm volatile("s_wait_asynccnt 0" ::: "memory");
    __syncthreads();

    // ---- Build packed {y0, slope} table (STRENGTH == 1.0 -> y = ran_y) ----
    if (tid < N_BINS - 1) {
        float ymin = rawY[0], ymax = rawY[0];
#pragma unroll
        for (int i = 1; i < N_BINS; ++i) {
            float v = rawY[i];
            ymin = fminf(ymin, v);
            ymax = fmaxf(ymax, v);
        }
        float s  = 1.0f / (ymax - ymin + EPS);
        float y0 = (rawY[tid]     - ymin) * s;
        float y1 = (rawY[tid + 1] - ymin) * s;
        lut[tid] = make_float2(y0, (y1 - y0) * (float)(N_BINS - 1));
    }
    __syncthreads();

    // ---- Streaming LUT application: b128 NT loads/stores, 8 in flight ----
    const v4f* __restrict__ xin = (const v4f*)x   + (size_t)b * vol4;
    v4f*       __restrict__ xo  = (v4f*)      out + (size_t)b * vol4;

    const int stride = gridDim.x * blockDim.x;        // in float4 units
    const int i0     = blockIdx.x * blockDim.x + tid;

    // Uniform (scalar) test: does this block's entire 8-iteration tile fit?
    const bool full = (blockIdx.x * blockDim.x + THREADS + (ITERS - 1) * stride)
                      <= vol4;

    if (full) {
        // Fast path: no per-lane guards -> all 8 b128 loads issue back-to-back.
        v4f v[ITERS];
#pragma unroll
        for (int it = 0; it < ITERS; ++it)
            v[it] = __builtin_nontemporal_load(&xin[i0 + it * stride]);
#pragma unroll
        for (int it = 0; it < ITERS; ++it)
            __builtin_nontemporal_store(apply4(v[it], lut), &xo[i0 + it * stride]);
    } else {
        // Remainder path (not taken for the 8x192^3 shape, kept for generality).
        for (int i = i0; i < vol4; i += stride) {
            v4f v = __builtin_nontemporal_load(&xin[i]);
            __builtin_nontemporal_store(apply4(v, lut), &xo[i]);
        }
    }
}

extern "C" void kernel_launch(void* const* d_in, const int* in_sizes, int n_in,
                              void* d_out, int out_size, void* d_ws, size_t ws_size,
                              hipStream_t stream) {
    const float* x     = (const float*)d_in[0];
    const float* ran_y = (const float*)d_in[1];
    float*       out   = (float*)d_out;

    const int bs   = in_sizes[1] / N_BINS;                 // 8
    const long long vol = (long long)in_sizes[0] / bs;     // 192^3 = 7,077,888
    const int vol4 = (int)(vol / 4);                       // 1,769,472

    const int gx = (vol4 + THREADS * ITERS - 1) / (THREADS * ITERS);  // 864
    dim3 grid(gx, bs);
    AugLUT_36455682408915_kernel<<<grid, THREADS, 0, stream>>>(x, ran_y, out, vol4);
}